// KalmanFilter_34273839022505
// MI455X (gfx1250) — compile-verified
//
#include <hip/hip_runtime.h>
#include <hip/hip_bf16.h>

typedef float v2f __attribute__((ext_vector_type(2)));
typedef float v8f __attribute__((ext_vector_type(8)));
typedef unsigned int u32x4 __attribute__((ext_vector_type(4)));
typedef int i32x4 __attribute__((ext_vector_type(4)));
typedef int i32x8 __attribute__((ext_vector_type(8)));

#define STATE   16
#define OBS     96
#define TLEN    8192
#define CHUNK   16
#define NCHUNK  (TLEN / CHUNK)

#define RICCATI_ITERS 40
#define NS_FIRST      12
#define NS_WARM       3

// ---------------------------------------------------------------------------
// WMMA helpers: V_WMMA_F32_16X16X4_F32, D(16x16) = A(16x4) @ B(4x16) + C
//   A frag: lane<16 -> (A[m][k0], A[m][k0+1]), lane>=16 -> (A[m][k0+2], A[m][k0+3])
//   B frag: v0 = B[k0+2*hi][n], v1 = B[k0+2*hi+1][n], n = lane&15
//   C/D:    c[v] = C[v + 8*hi][n]
// ---------------------------------------------------------------------------
static __device__ inline v8f wmma4(v2f a, v2f b, v8f c) {
  return __builtin_amdgcn_wmma_f32_16x16x4_f32(false, a, false, b, (short)0, c,
                                               false, false);
}

// acc += A(16xK, row-major lda) @ B(Kx16, row-major ldb). 1-stage pipelined.
static __device__ inline v8f mm_acc(const float* A, int lda, const float* B,
                                    int ldb, int K, v8f acc, int lane) {
  const int lo = lane & 15, hi = lane >> 4;
  const float* ap  = A + lo * lda + 2 * hi;
  const float* bp0 = B + (2 * hi) * ldb + lo;
  const float* bp1 = bp0 + ldb;
  v2f a, b;
  a.x = ap[0];  a.y = ap[1];
  b.x = bp0[0]; b.y = bp1[0];
  for (int k0 = 4; k0 < K; k0 += 4) {
    v2f an, bn;                         // prefetch next frags, then issue WMMA
    an.x = ap[k0];        an.y = ap[k0 + 1];
    bn.x = bp0[k0 * ldb]; bn.y = bp1[k0 * ldb];
    acc = wmma4(a, b, acc);
    a = an; b = bn;
  }
  return wmma4(a, b, acc);
}

// acc += A(16xK, lda) @ X^T  (X: 16 rows x K cols, row-major ldx). Pipelined.
static __device__ inline v8f mm_acc_bt(const float* A, int lda, const float* X,
                                       int ldx, int K, v8f acc, int lane) {
  const int lo = lane & 15, hi = lane >> 4;
  const float* ap = A + lo * lda + 2 * hi;
  const float* xp = X + lo * ldx + 2 * hi;
  v2f a, b;
  a.x = ap[0]; a.y = ap[1];
  b.x = xp[0]; b.y = xp[1];
  for (int k0 = 4; k0 < K; k0 += 4) {
    v2f an, bn;
    an.x = ap[k0]; an.y = ap[k0 + 1];
    bn.x = xp[k0]; bn.y = xp[k0 + 1];
    acc = wmma4(a, b, acc);
    a = an; b = bn;
  }
  return wmma4(a, b, acc);
}

static __device__ inline void store_tile(float* D, int ldd, v8f acc, int lane) {
  const int lo = lane & 15, hi = lane >> 4;
#pragma unroll
  for (int v = 0; v < 8; ++v) D[(v + 8 * hi) * ldd + lo] = acc[v];
}

static __device__ inline v8f load_tile(const float* Cc, int ldc, int lane) {
  const int lo = lane & 15, hi = lane >> 4;
  v8f c;
#pragma unroll
  for (int v = 0; v < 8; ++v) c[v] = Cc[(v + 8 * hi) * ldc + lo];
  return c;
}

// ---------------------------------------------------------------------------
// TDM: async 2D tile load global -> LDS (CDNA5 Tensor Data Mover).
// Builds D# group0/group1 per ISA 8.3/8.4: count=1, type=2, data_size=4B,
// tensor == tile (dim0 x dim1), row stride = stride0 elements.
// ---------------------------------------------------------------------------
static __device__ inline void tdm_load_2d_f32(unsigned lds_addr,
                                              const void* gaddr, unsigned dim0,
                                              unsigned dim1, unsigned stride0) {
  const unsigned long long ga = (unsigned long long)(size_t)gaddr;
  u32x4 g0;
  g0[0] = 1u;                                         // count=1, user mode
  g0[1] = lds_addr;                                   // lds_addr (bytes)
  g0[2] = (unsigned)(ga & 0xFFFFFFFFu);               // global_addr[31:0]
  g0[3] = (unsigned)((ga >> 32) & 0x01FFFFFFu) | 0x80000000u;  // [56:32]|type=2
  i32x8 g1;
  g1[0] = (int)(2u << 16);                            // data_size=4B, mask=0
  g1[1] = (int)((dim0 & 0xFFFFu) << 16);              // tensor_dim0[15:0]
  g1[2] = (int)(((dim0 >> 16) & 0xFFFFu) | ((dim1 & 0xFFFFu) << 16));
  g1[3] = (int)(((dim1 >> 16) & 0xFFFFu) | ((dim0 & 0xFFFFu) << 16)); // tile0
  g1[4] = (int)(dim1 & 0xFFFFu);                      // tile_dim1, tile_dim2=0
  g1[5] = (int)stride0;                               // tensor_dim0_stride
  g1[6] = 0;
  g1[7] = 0;
  i32x4 gz = {0, 0, 0, 0};
#if __clang_major__ >= 23
  i32x8 gz8 = {0, 0, 0, 0, 0, 0, 0, 0};
  __builtin_amdgcn_tensor_load_to_lds(g0, g1, gz, gz, gz8, 0);
#else
  __builtin_amdgcn_tensor_load_to_lds(g0, g1, gz, gz, 0);
#endif
}

// ---------------------------------------------------------------------------
// Kernel 1: steady-state Riccati iteration. One block (256 thr = 8 waves),
// everything in LDS.  Produces in ws:  K (16x96),  W_p=(M^T)^p p=0..16.
// S^-1 via Newton-Schulz  X <- X(2I - S X), warm-started across iterations.
// C and R staged into LDS once via the Tensor Data Mover.
// ---------------------------------------------------------------------------
__global__ void kf_riccati_kernel(const float* __restrict__ Ag,
                                  const float* __restrict__ Cg,
                                  const float* __restrict__ Qg,
                                  const float* __restrict__ Rg,
                                  const float* __restrict__ Pinit,
                                  float* __restrict__ Kws,
                                  float* __restrict__ Wws) {
  extern __shared__ float smem[];
  float* sC   = smem;          // 96*16
  float* sP   = sC + 1536;     // 16*16
  float* sPp  = sP + 256;      // P_pred
  float* sAP  = sPp + 256;     // A@P  (later reused as W chain)
  float* sCP  = sAP + 256;     // 96*16
  float* sPCt = sCP + 1536;    // 16*96
  float* sK   = sPCt + 1536;   // 16*96
  float* sT0  = sK + 1536;     // 16*16 (KC / IKC / M^T)
  float* sT1  = sT0 + 256;     // 16*16 (M)
  float* sRed = sT1 + 256;     // 128 reduction scratch
  float* sS   = sRed + 128;    // 96*96
  float* sXa  = sS + 9216;     // 96*96
  float* sXb  = sXa + 9216;    // 96*96
  float* sY   = sXb + 9216;    // 96*96
  float* sR   = sY + 9216;     // 96*96 (R staged by TDM)

  const int tid = threadIdx.x;
  const int wave = tid >> 5;
  const int lane = tid & 31;

  // ---- TDM: stage C (96x16) and R (96x96) into LDS (async, wave 0) ----
  if (wave == 0) {
    tdm_load_2d_f32((unsigned)(size_t)sC, (const void*)Cg, 16u, 96u, 16u);
    tdm_load_2d_f32((unsigned)(size_t)sR, (const void*)Rg, 96u, 96u, 96u);
    __builtin_amdgcn_s_wait_tensorcnt(0);
  }
  if (tid < 256) sP[tid] = Pinit[tid];
  __syncthreads();

  float* Xcur = sXa;
  float* Xalt = sXb;

  for (int iter = 0; iter < RICCATI_ITERS; ++iter) {
    // ---- predict: P_pred = A P A^T + Q ----
    if (wave == 0) {
      v8f acc = {};
      acc = mm_acc(Ag, 16, sP, 16, 16, acc, lane);
      store_tile(sAP, 16, acc, lane);
    }
    __syncthreads();
    if (wave == 0) {
      v8f acc = load_tile(Qg, 16, lane);
      acc = mm_acc_bt(sAP, 16, Ag, 16, 16, acc, lane);
      store_tile(sPp, 16, acc, lane);
    }
    __syncthreads();

    // ---- CP = C @ P_pred (96x16), 6 row tiles ----
    if (wave < 6) {
      v8f acc = {};
      acc = mm_acc(sC + wave * 256, 16, sPp, 16, 16, acc, lane);
      store_tile(sCP + wave * 256, 16, acc, lane);
    }
    __syncthreads();

    // ---- S = CP @ C^T + R (96x96), 36 tiles ----
    for (int t = wave; t < 36; t += 8) {
      const int bm = t / 6, bn = t % 6;
      v8f acc = load_tile(sR + bm * 16 * 96 + bn * 16, 96, lane);
      acc = mm_acc_bt(sCP + bm * 256, 16, sC + bn * 256, 16, 16, acc, lane);
      store_tile(sS + bm * 16 * 96 + bn * 16, 96, acc, lane);
    }
    __syncthreads();

    // ---- S^-1 via Newton-Schulz ----
    if (iter == 0) {
      if (tid < 96) {
        float s = 0.f;
        for (int j = 0; j < 96; ++j) s += fabsf(sS[tid * 96 + j]);
        sRed[tid] = s;
      }
      __syncthreads();
      if (tid == 0) {
        float m = 0.f;
        for (int j = 0; j < 96; ++j) m = fmaxf(m, sRed[j]);
        sRed[96] = 1.0f / (m * m);
      }
      __syncthreads();
      const float inv = sRed[96];
      for (int e = tid; e < 9216; e += 256) {
        const int r = e / 96, c = e % 96;
        Xcur[e] = sS[c * 96 + r] * inv;  // X0 = S^T / (|S|_1 |S|_inf)
      }
      __syncthreads();
    }
    const int nns = (iter == 0) ? NS_FIRST : NS_WARM;
    for (int ns = 0; ns < nns; ++ns) {
      for (int t = wave; t < 36; t += 8) {             // Y = S @ X
        const int bm = t / 6, bn = t % 6;
        v8f acc = {};
        acc = mm_acc(sS + bm * 16 * 96, 96, Xcur + bn * 16, 96, 96, acc, lane);
        store_tile(sY + bm * 16 * 96 + bn * 16, 96, acc, lane);
      }
      __syncthreads();
      for (int e = tid; e < 9216; e += 256) {          // Z = 2I - Y
        const int r = e / 96, c = e % 96;
        float v = -sY[e];
        if (r == c) v += 2.0f;
        sY[e] = v;
      }
      __syncthreads();
      for (int t = wave; t < 36; t += 8) {             // Xnew = X @ Z
        const int bm = t / 6, bn = t % 6;
        v8f acc = {};
        acc = mm_acc(Xcur + bm * 16 * 96, 96, sY + bn * 16, 96, 96, acc, lane);
        store_tile(Xalt + bm * 16 * 96 + bn * 16, 96, acc, lane);
      }
      __syncthreads();
      float* tmp = Xcur; Xcur = Xalt; Xalt = tmp;
    }

    // ---- PCt = P_pred @ C^T (16x96) ----
    if (wave < 6) {
      v8f acc = {};
      acc = mm_acc_bt(sPp, 16, sC + wave * 256, 16, 16, acc, lane);
      store_tile(sPCt + wave * 16, 96, acc, lane);
    }
    __syncthreads();

    // ---- K = PCt @ Sinv (16x96, K-dim 96) ----
    if (wave < 6) {
      v8f acc = {};
      acc = mm_acc(sPCt, 96, Xcur + wave * 16, 96, 96, acc, lane);
      store_tile(sK + wave * 16, 96, acc, lane);
    }
    __syncthreads();

    // ---- KC = K @ C (16x16, K-dim 96); IKC = I - KC ----
    if (wave == 0) {
      v8f acc = {};
      acc = mm_acc(sK, 96, sC, 16, 96, acc, lane);
      store_tile(sT0, 16, acc, lane);
    }
    __syncthreads();
    {
      const int r = tid / 16, c = tid % 16;
      float v = -sT0[tid];
      if (r == c) v += 1.0f;
      __syncthreads();
      sT0[tid] = v;
    }
    __syncthreads();

    // ---- P = IKC @ P_pred ;  M = IKC @ A ----
    if (wave == 0) {
      v8f acc = {};
      acc = mm_acc(sT0, 16, sPp, 16, 16, acc, lane);
      store_tile(sP, 16, acc, lane);
    } else if (wave == 1) {
      v8f acc = {};
      acc = mm_acc(sT0, 16, Ag, 16, 16, acc, lane);
      store_tile(sT1, 16, acc, lane);
    }
    __syncthreads();
  }

  // ---- epilogue: export K and W_p = (M^T)^p, p = 0..16 ----
  for (int e = tid; e < 1536; e += 256) Kws[e] = sK[e];
  {
    const int i = tid / 16, j = tid % 16;
    sT0[i * 16 + j] = sT1[j * 16 + i];          // M^T
    const float id = (i == j) ? 1.0f : 0.0f;    // W_0 = I
    sAP[tid] = id;
    Wws[tid] = id;
  }
  __syncthreads();
  for (int p = 1; p <= 16; ++p) {
    if (wave == 0) {
      v8f acc = {};
      acc = mm_acc(sAP, 16, sT0, 16, 16, acc, lane);
      store_tile(sAP, 16, acc, lane);
      store_tile(Wws + p * 256, 16, acc, lane);
    }
    __syncthreads();
  }
}

// ---------------------------------------------------------------------------
// Kernel 2 (bulk, fused): per 16-step chunk (one wave each):
//   Btile = Y[16x96] @ K^T               (24 WMMAs)
//   U     = sum_p Shift_p(Btile) @ W_p   (64 WMMAs, causal chunk-local scan)
// K and the W stack staged once per block into LDS.
// ---------------------------------------------------------------------------
__global__ void kf_gain_scan_kernel(const float* __restrict__ Yg,
                                    const float* __restrict__ Kg,
                                    const float* __restrict__ Wg,
                                    float* __restrict__ out) {
  __shared__ float sK[1536];
  __shared__ float sW[17 * 256];
  __shared__ float sB[8][256];
  const int tid = threadIdx.x;
  const int wave = tid >> 5;
  const int lane = tid & 31;
  const int lo = lane & 15, hi = lane >> 4;
  const int chunk = blockIdx.x * 8 + wave;
  const int t0 = chunk * CHUNK;

  for (int e = tid; e < 1536; e += 256) sK[e] = Kg[e];
  for (int e = tid; e < 17 * 256; e += 256) sW[e] = Wg[e];
  __builtin_prefetch(Yg + t0 * OBS, 0, 1);
  __syncthreads();

  // Btile = Y_chunk @ K^T
  v8f acc = {};
  acc = mm_acc_bt(Yg + t0 * OBS, OBS, sK, OBS, OBS, acc, lane);
  store_tile(sB[wave], 16, acc, lane);

  // U = sum_{p=0}^{15} Shift_down_p(B) @ (M^T)^p
  v8f acc2 = {};
  const float* Bt = sB[wave];
#pragma unroll 4
  for (int p = 0; p < 16; ++p) {
    const float* W = sW + p * 256;
    const int r = lo - p;
#pragma unroll
    for (int k0 = 0; k0 < 16; k0 += 4) {
      v2f a, b;
      a.x = (r >= 0) ? Bt[r * 16 + k0 + 2 * hi] : 0.f;
      a.y = (r >= 0) ? Bt[r * 16 + k0 + 2 * hi + 1] : 0.f;
      b.x = W[(k0 + 2 * hi) * 16 + lo];
      b.y = W[(k0 + 2 * hi + 1) * 16 + lo];
      acc2 = wmma4(a, b, acc2);
    }
  }
  store_tile(out + t0 * STATE, STATE, acc2, lane);
}

// ---------------------------------------------------------------------------
// Kernel 3: carry scan across the 512 chunks (tiny): s_{c+1} = s_c W16 + U_c[15]
// ---------------------------------------------------------------------------
__global__ void kf_carry_kernel(const float* __restrict__ xinit,
                                const float* __restrict__ Wg,
                                const float* __restrict__ U,
                                float* __restrict__ Svec) {
  __shared__ float r[16];
  const int n = threadIdx.x;  // blockDim = 16, single wave
  const float* W16 = Wg + 16 * 256;
  r[n] = xinit[n];
  __syncthreads();
  for (int c = 0; c < NCHUNK; ++c) {
    Svec[c * 16 + n] = r[n];
    float acc = U[(c * 16 + 15) * STATE + n];
#pragma unroll
    for (int k = 0; k < 16; ++k) acc += r[k] * W16[k * 16 + n];
    __syncthreads();
    r[n] = acc;
    __syncthreads();
  }
}

// ---------------------------------------------------------------------------
// Kernel 4: recombine  x[c*16+j] = U[c*16+j] + s_c (M^T)^{j+1}
// ---------------------------------------------------------------------------
__global__ void kf_finalize_kernel(const float* __restrict__ Svec,
                                   const float* __restrict__ Wg,
                                   float* __restrict__ out) {
  __shared__ float sv[16];
  const int c = blockIdx.x;
  const int j = threadIdx.x >> 4;
  const int n = threadIdx.x & 15;
  if (threadIdx.x < 16) sv[threadIdx.x] = Svec[c * 16 + threadIdx.x];
  __syncthreads();
  const float* Wj = Wg + (j + 1) * 256;
  float acc = out[(c * 16 + j) * STATE + n];
#pragma unroll
  for (int k = 0; k < 16; ++k) acc += sv[k] * Wj[k * 16 + n];
  out[(c * 16 + j) * STATE + n] = acc;
}

// ---------------------------------------------------------------------------
extern "C" void kernel_launch(void* const* d_in, const int* in_sizes, int n_in,
                              void* d_out, int out_size, void* d_ws,
                              size_t ws_size, hipStream_t stream) {
  (void)in_sizes; (void)n_in; (void)out_size; (void)ws_size;
  const float* Yg    = (const float*)d_in[0];  // y_seq [8192,96]
  const float* Ag    = (const float*)d_in[1];  // A [16,16]
  const float* Cg    = (const float*)d_in[2];  // C [96,16]
  const float* Qg    = (const float*)d_in[3];  // Q [16,16]
  const float* Rg    = (const float*)d_in[4];  // R [96,96]
  const float* xinit = (const float*)d_in[5];  // x_init [16]
  const float* Pinit = (const float*)d_in[6];  // P_init [16,16]
  float* out = (float*)d_out;

  float* wsf  = (float*)d_ws;
  float* Kws  = wsf;                 // 16*96           = 1536
  float* Wws  = wsf + 1536;          // 17 * 16*16      = 4352
  float* Svec = wsf + 1536 + 4352;   // NCHUNK * 16     = 8192   (~56 KB total)

  const size_t smem_bytes = 53632u * sizeof(float);  // ~210 KB of 320 KB WGP LDS
  (void)hipFuncSetAttribute((const void*)kf_riccati_kernel,
                            hipFuncAttributeMaxDynamicSharedMemorySize,
                            (int)smem_bytes);

  kf_riccati_kernel<<<1, 256, smem_bytes, stream>>>(Ag, Cg, Qg, Rg, Pinit,
                                                    Kws, Wws);
  kf_gain_scan_kernel<<<NCHUNK / 8, 256, 0, stream>>>(Yg, Kws, Wws, out);
  kf_carry_kernel<<<1, 16, 0, stream>>>(xinit, Wws, out, Svec);
  kf_finalize_kernel<<<NCHUNK, 256, 0, stream>>>(Svec, Wws, out);
}